// VGAE_15753940041956
// MI455X (gfx1250) — compile-verified
//
#include <hip/hip_runtime.h>
#include <math.h>

// ---------------------------------------------------------------------------
// VGAE forward for MI455X (gfx1250, wave32).
// fp32 WMMA (V_WMMA_F32_16X16X4_F32) for the dense node transforms,
// LDS-staged weights (K-pair swizzled so B fragments are single ds_load_b64),
// fused bias/ELU/deg-normalization epilogues,
// wave-per-edge float4 atomic scatter for the graph aggregations.
// ---------------------------------------------------------------------------

typedef float v2f __attribute__((ext_vector_type(2)));
typedef float v8f __attribute__((ext_vector_type(8)));

__global__ void fill_zero_kernel(float* __restrict__ p, int n) {
  int i = blockIdx.x * blockDim.x + threadIdx.x;
  if (i < n) p[i] = 0.0f;
}

__global__ void degree_kernel(const int* __restrict__ s, const int* __restrict__ r,
                              float* __restrict__ sdeg, float* __restrict__ rdeg, int E) {
  int i = blockIdx.x * blockDim.x + threadIdx.x;
  int stride = gridDim.x * blockDim.x;
  for (; i < E; i += stride) {
    atomicAdd(&sdeg[s[i]], 1.0f);
    atomicAdd(&rdeg[r[i]], 1.0f);
  }
}

// Four normalization vectors: with self-edge (+1) and without (max(deg,1)).
__global__ void isq_kernel(const float* __restrict__ sdeg, const float* __restrict__ rdeg,
                           float* __restrict__ issS, float* __restrict__ irsS,
                           float* __restrict__ isN, float* __restrict__ irN, int N) {
  int i = blockIdx.x * blockDim.x + threadIdx.x;
  if (i < N) {
    float sd = sdeg[i], rd = rdeg[i];
    issS[i] = rsqrtf(sd + 1.0f);          // self-edges: deg+1 >= 1 always
    irsS[i] = rsqrtf(rd + 1.0f);
    isN[i]  = rsqrtf(fmaxf(sd, 1.0f));    // no self-edges
    irN[i]  = rsqrtf(fmaxf(rd, 1.0f));
  }
}

// ---------------------------------------------------------------------------
// C[N,DOUT] = postproc(rsIn[row] * A[N,DIN] @ W[DIN,DOUT] + bias) * rsOut[row]
//   postproc = ELU (alpha=1) if DO_ELU
// Also optionally writes accInit[row,col] = accZero ? 0 : C value
// (seeds the scatter accumulator: self-edge contribution or zero).
// One wave -> 16-row strip x DOUT cols. fp32 WMMA 16x16x4.
//
// LDS weight layout is K-pair swizzled: ldsW[(k>>1)*DOUT*2 + col*2 + (k&1)],
// so the B fragment {W[kb][col], W[kb+1][col]} is one aligned 8-byte ds load.
// ---------------------------------------------------------------------------
template <int DIN, int DOUT, bool DO_ELU>
__global__ __launch_bounds__(256) void gemm_wmma_f32(
    const float* __restrict__ A, const float* __restrict__ W,
    const float* __restrict__ bias,
    const float* __restrict__ rsIn,   // may be null
    const float* __restrict__ rsOut,  // may be null
    float* __restrict__ out,
    float* __restrict__ accInit,      // may be null
    int accZero, int nRows) {
  constexpr int NT = DOUT / 16;
  __shared__ float ldsW[DIN * DOUT];

  const int tid = threadIdx.x;
  // Stage weights once per workgroup (<= 64KB), K-pair swizzled.
  for (int i = tid; i < DIN * DOUT; i += 256) {
    const int k = i / DOUT;
    const int col = i - k * DOUT;
    ldsW[((k >> 1) * DOUT + col) * 2 + (k & 1)] = W[i];
  }
  __syncthreads();

  const int wave = tid >> 5;
  const int lane = tid & 31;
  const int laneLo = lane & 15;
  const int laneHi = lane >> 4;  // 0 or 1
  const int stripRow0 = (blockIdx.x * 8 + wave) * 16;
  if (stripRow0 >= nRows) return;  // wave-uniform: EXEC stays all-ones for WMMA

  const int myRow = stripRow0 + laneLo;
  const int rdRow = myRow < nRows ? myRow : nRows - 1;  // clamp; store is masked
  const float rin = rsIn ? rsIn[rdRow] : 1.0f;
  const float* __restrict__ arow = A + (size_t)rdRow * DIN;

  v8f acc[NT];
  v8f vz = {};
#pragma unroll
  for (int t = 0; t < NT; ++t) acc[t] = vz;

  // A 16x4 fragment: lanes 0-15 hold K = k,k+1 ; lanes 16-31 hold K = k+2,k+3
#pragma unroll 4
  for (int k = 0; k < DIN; k += 4) {
    const int kb = k + 2 * laneHi;                       // even
    v2f a = *(const v2f*)(arow + kb);
    a *= rin;                                            // v_pk_mul_f32
    const float* __restrict__ brow = &ldsW[(size_t)(kb >> 1) * DOUT * 2];
#pragma unroll
    for (int t = 0; t < NT; ++t) {
      const int col = t * 16 + laneLo;
      const v2f b = *(const v2f*)(brow + col * 2);       // single ds_load_b64
      acc[t] = __builtin_amdgcn_wmma_f32_16x16x4_f32(
          /*neg_a=*/false, a, /*neg_b=*/false, b,
          /*c_mod=*/(short)0, acc[t], /*reuse_a=*/false, /*reuse_b=*/false);
    }
  }

  // C/D layout: lane col = n0 + laneLo; VGPR p -> row p (+8 for lanes 16-31)
#pragma unroll
  for (int t = 0; t < NT; ++t) {
    const int col = t * 16 + laneLo;
    const float bv = bias[col];
#pragma unroll
    for (int p = 0; p < 8; ++p) {
      const int row = stripRow0 + p + 8 * laneHi;
      if (row < nRows) {
        float v = acc[t][p] + bv;
        if (DO_ELU) v = v > 0.0f ? v : expm1f(v);
        if (rsOut) v *= rsOut[row];
        out[(size_t)row * DOUT + col] = v;
        if (accInit) accInit[(size_t)row * DOUT + col] = accZero ? 0.0f : v;
      }
    }
  }
}

// ---------------------------------------------------------------------------
// Scatter-sum: acc[r[e], :] += src[s[e], :].  One wave per edge, lanes carry
// feature columns (float4 per lane for D=128, float2 for D=64).
// ---------------------------------------------------------------------------
template <int D>
__global__ __launch_bounds__(256) void scatter_kernel(
    const float* __restrict__ src, float* __restrict__ acc,
    const int* __restrict__ s, const int* __restrict__ r, int E) {
  constexpr int F = D / 32;
  const int lane = threadIdx.x & 31;
  int e = (blockIdx.x * 256 + threadIdx.x) >> 5;
  const int nw = (gridDim.x * 256) >> 5;
  for (; e < E; e += nw) {
    const int se = s[e];
    const int re = r[e];
    const float* __restrict__ sp = src + (size_t)se * D + lane * F;
    float* __restrict__ dp = acc + (size_t)re * D + lane * F;
    if (F == 4) {
      const float4 v = *(const float4*)sp;
      atomicAdd(dp + 0, v.x);
      atomicAdd(dp + 1, v.y);
      atomicAdd(dp + 2, v.z);
      atomicAdd(dp + 3, v.w);
    } else {
      const float2 v = *(const float2*)sp;
      atomicAdd(dp + 0, v.x);
      atomicAdd(dp + 1, v.y);
    }
  }
}

// mean/log_std finalize + reparameterization z = mean + exp(log_std)*eps
__global__ void z_kernel(const float* __restrict__ accMu, const float* __restrict__ accLs,
                         const float* __restrict__ irN, const float* __restrict__ eps,
                         float* __restrict__ meanOut, float* __restrict__ lsOut,
                         float* __restrict__ z, int N) {
  int i = blockIdx.x * blockDim.x + threadIdx.x;
  const int total = N * 64;
  if (i < total) {
    const int row = i >> 6;
    const float sc = irN[row];
    const float m = accMu[i] * sc;
    const float l = accLs[i] * sc;
    meanOut[i] = m;
    lsOut[i] = l;
    z[i] = m + expf(l) * eps[i];
  }
}

__global__ void finish_kernel(const float* __restrict__ acc, const float* __restrict__ irN,
                              float* __restrict__ out, int N) {
  int i = blockIdx.x * blockDim.x + threadIdx.x;
  const int total = N * 128;
  if (i < total) out[i] = acc[i] * irN[i >> 7];
}

// ---------------------------------------------------------------------------

extern "C" void kernel_launch(void* const* d_in, const int* in_sizes, int n_in,
                              void* d_out, int out_size, void* d_ws, size_t ws_size,
                              hipStream_t stream) {
  const float* nodes = (const float*)d_in[0];
  const int* senders = (const int*)d_in[1];
  const int* receivers = (const int*)d_in[2];
  const float* eps = (const float*)d_in[3];
  const float* W_h = (const float*)d_in[4];
  const float* b_h = (const float*)d_in[5];
  const float* W_mu = (const float*)d_in[6];
  const float* b_mu = (const float*)d_in[7];
  const float* W_ls = (const float*)d_in[8];
  const float* b_ls = (const float*)d_in[9];
  const float* W_dh = (const float*)d_in[10];
  const float* b_dh = (const float*)d_in[11];
  const float* W_do = (const float*)d_in[12];
  const float* b_do = (const float*)d_in[13];

  const int N = in_sizes[0] / 128;
  const int E = in_sizes[1];

  float* mean_out = (float*)d_out;                    // [N,64]
  float* ls_out = mean_out + (size_t)N * 64;          // [N,64]
  float* out_out = mean_out + (size_t)N * 128;        // [N,128]

  // Workspace carve-out (floats)
  float* ws = (float*)d_ws;
  float* sdeg = ws;                      // N   (sdeg+rdeg contiguous -> one fill)
  float* rdeg = sdeg + N;                // N
  float* issS = rdeg + N;                // N  rsqrt(sdeg+1)
  float* irsS = issS + N;                // N  rsqrt(rdeg+1)
  float* isN = irsS + N;                 // N  rsqrt(max(sdeg,1))
  float* irN = isN + N;                  // N  rsqrt(max(rdeg,1))
  float* bufA = irN + N;                 // N*128  scaled features (Xs1 / Xs4 / Po)
  float* bufB = bufA + (size_t)N * 128;  // N*128  accumulator (Acc1 / Acc4)
  float* bufC = bufB + (size_t)N * 128;  // N*128  accumulator Acc5
  float* bufD = bufC + (size_t)N * 128;  // N*64   Pmu / Pls / z
  float* bufE = bufD + (size_t)N * 64;   // N*64   AccMu
  float* bufF = bufE + (size_t)N * 64;   // N*64   AccLs

  const int T = 256;
  const int gB = (N + 127) / 128;        // GEMM blocks: 8 waves x 16 rows
  const int gScat = 2048;                // 16k waves grid-striding over edges

  // --- degrees & normalization vectors ---
  fill_zero_kernel<<<(2 * N + T - 1) / T, T, 0, stream>>>(sdeg, 2 * N);
  degree_kernel<<<1024, T, 0, stream>>>(senders, receivers, sdeg, rdeg, E);
  isq_kernel<<<(N + T - 1) / T, T, 0, stream>>>(sdeg, rdeg, issS, irsS, isN, irN, N);

  // --- encoder hidden: h = elu(X@W_h+b_h); agg with self-edges ---
  gemm_wmma_f32<128, 128, true><<<gB, T, 0, stream>>>(
      nodes, W_h, b_h, nullptr, issS, bufA, bufB, /*accZero=*/0, N);
  scatter_kernel<128><<<gScat, T, 0, stream>>>(bufA, bufB, senders, receivers, E);

  // --- mean: (h_agg*irsS)@W_mu+b_mu scaled isN; agg no self ---
  gemm_wmma_f32<128, 64, false><<<gB, T, 0, stream>>>(
      bufB, W_mu, b_mu, irsS, isN, bufD, bufE, /*accZero=*/1, N);
  scatter_kernel<64><<<gScat, T, 0, stream>>>(bufD, bufE, senders, receivers, E);

  // --- log_std ---
  gemm_wmma_f32<128, 64, false><<<gB, T, 0, stream>>>(
      bufB, W_ls, b_ls, irsS, isN, bufD, bufF, /*accZero=*/1, N);
  scatter_kernel<64><<<gScat, T, 0, stream>>>(bufD, bufF, senders, receivers, E);

  // --- finalize mean/log_std, reparameterize z ---
  z_kernel<<<((size_t)N * 64 + T - 1) / T, T, 0, stream>>>(
      bufE, bufF, irN, eps, mean_out, ls_out, bufD, N);

  // --- decoder hidden: d = elu(z@W_dh+b_dh); agg with self-edges ---
  gemm_wmma_f32<64, 128, true><<<gB, T, 0, stream>>>(
      bufD, W_dh, b_dh, nullptr, issS, bufA, bufB, /*accZero=*/0, N);
  scatter_kernel<128><<<gScat, T, 0, stream>>>(bufA, bufB, senders, receivers, E);

  // --- decoder out: (d_agg*irsS)@W_do+b_do scaled isN; agg no self ---
  gemm_wmma_f32<128, 128, false><<<gB, T, 0, stream>>>(
      bufB, W_do, b_do, irsS, isN, bufA, bufC, /*accZero=*/1, N);
  scatter_kernel<128><<<gScat, T, 0, stream>>>(bufA, bufC, senders, receivers, E);

  finish_kernel<<<((size_t)N * 128 + T - 1) / T, T, 0, stream>>>(bufC, irN, out_out, N);
}